// APPNPModel_28587302322286
// MI455X (gfx1250) — compile-verified
//
#include <hip/hip_runtime.h>
#include <hip/hip_bf16.h>

// APPNP forward for MI455X (gfx1250), wave32.
// Hot loop: propagator[16384x16384] fp32 @ z[16384x16] via V_WMMA_F32_16X16X4_F32.
// HBM-bound at 1 GiB propagator read (~46us @ 23.3 TB/s); fp32 WMMA keeps
// exact math while the matrix pipe paces HBM.

typedef __attribute__((ext_vector_type(2))) float v2f;
typedef __attribute__((ext_vector_type(8))) float v8f;

#define N_NODES     16384
#define N_FEATURES  1024
#define NNZ_CNT     819200
#define HIDDEN      64
#define N_LABELS    16
#define KSPLIT      16          // K partitions for propagator GEMM
#define RBLK        4           // 16-row tiles per wave in propagator GEMM

// ---------------------------------------------------------------- utilities
__global__ void k_zero(float4* __restrict__ p, int n4) {
    int i = blockIdx.x * blockDim.x + threadIdx.x;
    int stride = gridDim.x * blockDim.x;
    for (; i < n4; i += stride) p[i] = make_float4(0.f, 0.f, 0.f, 0.f);
}

__global__ void k_scatter(const int* __restrict__ rows, const int* __restrict__ cols,
                          const float* __restrict__ vals, float* __restrict__ Xd) {
    int t = blockIdx.x * blockDim.x + threadIdx.x;
    if (t < NNZ_CNT) {
        atomicAdd(&Xd[rows[t] * N_FEATURES + cols[t]], vals[t]);
    }
}

// --------------------------------------------------- GEMM1: h = relu(Xd@W1+b1)
// 1024 row-blocks x 4 col-blocks = 4096 waves (512 blocks x 8 waves).
__global__ void __launch_bounds__(256)
k_gemm1(const float* __restrict__ Xd, const float* __restrict__ W1,
        const float* __restrict__ b1, float* __restrict__ h) {
    const int lane  = threadIdx.x & 31;
    const int wid   = blockIdx.x * 8 + (threadIdx.x >> 5);
    const int rb    = wid >> 2;                 // 0..1023
    const int cb    = wid & 3;                  // 0..3
    const int khalf = (lane >> 4) << 1;         // lanes 0-15 -> K{0,1}; 16-31 -> K{2,3}
    const int col   = (cb << 4) + (lane & 15);

    v8f c = {};
    const float* arow = Xd + (long)(rb * 16 + (lane & 15)) * N_FEATURES;
    for (int k0 = 0; k0 < N_FEATURES; k0 += 4) {
        v2f a = *(const v2f*)(arow + k0 + khalf);
        v2f b;
        b.x = W1[(k0 + khalf)     * HIDDEN + col];
        b.y = W1[(k0 + khalf + 1) * HIDDEN + col];
        c = __builtin_amdgcn_wmma_f32_16x16x4_f32(false, a, false, b,
                                                  (short)0, c, false, false);
    }
    const float bias  = b1[col];
    const int   mbase = rb * 16 + ((lane >> 4) << 3); // C layout: vgpr v -> row v (+8 hi-half)
    #pragma unroll
    for (int v = 0; v < 8; ++v) {
        float val = c[v] + bias;
        h[(mbase + v) * HIDDEN + col] = val > 0.f ? val : 0.f;
    }
}

// --------------------------------------------------- GEMM2: z = h@W2 + b2
// 1024 waves (128 blocks x 8 waves), single 16-wide col block.
__global__ void __launch_bounds__(256)
k_gemm2(const float* __restrict__ h, const float* __restrict__ W2,
        const float* __restrict__ b2, float* __restrict__ z) {
    const int lane  = threadIdx.x & 31;
    const int rb    = blockIdx.x * 8 + (threadIdx.x >> 5); // 0..1023
    const int khalf = (lane >> 4) << 1;
    const int col   = lane & 15;

    v8f c = {};
    const float* arow = h + (long)(rb * 16 + (lane & 15)) * HIDDEN;
    #pragma unroll
    for (int k0 = 0; k0 < HIDDEN; k0 += 4) {
        v2f a = *(const v2f*)(arow + k0 + khalf);
        v2f b;
        b.x = W2[(k0 + khalf)     * N_LABELS + col];
        b.y = W2[(k0 + khalf + 1) * N_LABELS + col];
        c = __builtin_amdgcn_wmma_f32_16x16x4_f32(false, a, false, b,
                                                  (short)0, c, false, false);
    }
    const float bias  = b2[col];
    const int   mbase = rb * 16 + ((lane >> 4) << 3);
    #pragma unroll
    for (int v = 0; v < 8; ++v) {
        z[(mbase + v) * N_LABELS + col] = c[v] + bias;
    }
}

// --------------------------------- propagator GEMM: partial[s] = prop_ks @ z_ks
// 4096 waves: 256 row-groups (64 rows each) x 16 K-partitions.
// Waves in one block share the same K partition 's' -> shared z cachelines.
__global__ void __launch_bounds__(256)
k_prop(const float* __restrict__ prop, const float* __restrict__ z,
       float* __restrict__ partial) {
    const int lane  = threadIdx.x & 31;
    const int wid   = blockIdx.x * 8 + (threadIdx.x >> 5); // 0..4095
    const int s     = wid >> 8;                 // 0..15 (same for 8 waves of a block)
    const int rg    = wid & 255;                // 0..255
    const int kbase = s * (N_NODES / KSPLIT);   // 1024-wide K window
    const int khalf = (lane >> 4) << 1;
    const int col   = lane & 15;
    const int row0  = rg * 64 + (lane & 15);

    const float* a0 = prop + (long)(row0)      * N_NODES;
    const float* a1 = prop + (long)(row0 + 16) * N_NODES;
    const float* a2 = prop + (long)(row0 + 32) * N_NODES;
    const float* a3 = prop + (long)(row0 + 48) * N_NODES;

    v8f c0 = {}, c1 = {}, c2 = {}, c3 = {};
    for (int kk = 0; kk < N_NODES / KSPLIT; kk += 4) {
        const int k = kbase + kk + khalf;
        v2f b;
        b.x = z[(k)     * N_LABELS + col];
        b.y = z[(k + 1) * N_LABELS + col];
        v2f A0 = *(const v2f*)(a0 + k);
        v2f A1 = *(const v2f*)(a1 + k);
        v2f A2 = *(const v2f*)(a2 + k);
        v2f A3 = *(const v2f*)(a3 + k);
        c0 = __builtin_amdgcn_wmma_f32_16x16x4_f32(false, A0, false, b, (short)0, c0, false, false);
        c1 = __builtin_amdgcn_wmma_f32_16x16x4_f32(false, A1, false, b, (short)0, c1, false, false);
        c2 = __builtin_amdgcn_wmma_f32_16x16x4_f32(false, A2, false, b, (short)0, c2, false, false);
        c3 = __builtin_amdgcn_wmma_f32_16x16x4_f32(false, A3, false, b, (short)0, c3, false, false);
    }

    float* pout = partial + (long)s * N_NODES * N_LABELS;
    const int mhalf = (lane >> 4) << 3;
    #pragma unroll
    for (int v = 0; v < 8; ++v) {
        pout[(rg * 64 +      mhalf + v) * N_LABELS + col] = c0[v];
        pout[(rg * 64 + 16 + mhalf + v) * N_LABELS + col] = c1[v];
        pout[(rg * 64 + 32 + mhalf + v) * N_LABELS + col] = c2[v];
        pout[(rg * 64 + 48 + mhalf + v) * N_LABELS + col] = c3[v];
    }
}

// --------------------------- reduce K partials + log_softmax over 16 labels
__global__ void __launch_bounds__(256)
k_softmax(const float* __restrict__ partial, float* __restrict__ out) {
    __shared__ float sm[16][17];
    const int j    = threadIdx.x & 15;          // label
    const int il   = threadIdx.x >> 4;          // local node 0..15
    const int node = blockIdx.x * 16 + il;

    float s = 0.f;
    #pragma unroll
    for (int p = 0; p < KSPLIT; ++p)
        s += partial[(long)p * N_NODES * N_LABELS + node * N_LABELS + j];
    sm[il][j] = s;
    __syncthreads();

    float m = sm[il][0];
    #pragma unroll
    for (int t = 1; t < 16; ++t) m = fmaxf(m, sm[il][t]);
    float sum = 0.f;
    #pragma unroll
    for (int t = 0; t < 16; ++t) sum += expf(sm[il][t] - m);
    out[node * N_LABELS + j] = s - m - logf(sum);
}

// ---------------------------------------------------------------- launcher
extern "C" void kernel_launch(void* const* d_in, const int* in_sizes, int n_in,
                              void* d_out, int out_size, void* d_ws, size_t ws_size,
                              hipStream_t stream) {
    const int*   fidx = (const int*)d_in[0];    // [2, NNZ]: rows then cols
    const float* fval = (const float*)d_in[1];
    const float* W1   = (const float*)d_in[2];
    const float* b1   = (const float*)d_in[3];
    const float* W2   = (const float*)d_in[4];
    const float* b2   = (const float*)d_in[5];
    const float* prop = (const float*)d_in[6];
    float*       out  = (float*)d_out;

    char*  ws = (char*)d_ws;
    size_t off = 0;
    float* Xd = (float*)(ws + off); off += (size_t)N_NODES * N_FEATURES * sizeof(float); // 64 MB
    float* h  = (float*)(ws + off); off += (size_t)N_NODES * HIDDEN     * sizeof(float); //  4 MB
    float* z  = (float*)(ws + off); off += (size_t)N_NODES * N_LABELS   * sizeof(float); //  1 MB
    float* partial = (float*)(ws + off); // KSPLIT * 1 MB = 16 MB

    const int* rows = fidx;
    const int* cols = fidx + NNZ_CNT;

    k_zero<<<4096, 256, 0, stream>>>((float4*)Xd, N_NODES * N_FEATURES / 4);
    k_scatter<<<(NNZ_CNT + 255) / 256, 256, 0, stream>>>(rows, cols, fval, Xd);
    k_gemm1<<<512, 256, 0, stream>>>(Xd, W1, b1, h);       // 4096 waves
    k_gemm2<<<128, 256, 0, stream>>>(h, W2, b2, z);        // 1024 waves
    k_prop <<<512, 256, 0, stream>>>(prop, z, partial);    // 4096 waves
    k_softmax<<<N_NODES / 16, 256, 0, stream>>>(partial, out);
}